// DecisionBoundary_78469052498221
// MI455X (gfx1250) — compile-verified
//
#include <hip/hip_runtime.h>

// DecisionBoundary: out[b] = 1 + max_{j!=y} softmax(x)_j - softmax(x)_y
// B = 262144 rows, C = 1000 classes, fp32 in/out.
//
// Memory-bound: 1.05 GB streamed once -> ~45us floor at 23.3 TB/s.
// Strategy: one wave32 per row. 1000 floats = 250 float4; lane L loads
// float4 indices {L, 32+L, ..., 224+L} (8 slots, last partially masked with
// -inf padding). All 8 b128 non-temporal loads are issued before first use
// so LOADcnt keeps the pipe full. Per-lane: local max, then one v_exp_f32
// per element for the exp-sum, tracking true-class logit (xy) and max wrong
// logit (mw). 5-step __shfl_xor wave reduction merges (m,s) with the
// online-softmax rescale identity and maxes mw/xy. Lane 0 writes the result.

typedef float v4f __attribute__((ext_vector_type(4)));

#define NUM_CLASS 1000
#define N_F4 250                 // 1000 / 4 float4 per row (4000 B, 16B aligned)
#define WAVES_PER_BLOCK 8
#define BLOCK 256

__global__ __launch_bounds__(BLOCK) void DecisionBoundary_kernel(
    const float* __restrict__ logits,
    const int*   __restrict__ y,
    float*       __restrict__ out,
    int batch)
{
    const int lane = threadIdx.x & 31;
    const int row  = blockIdx.x * WAVES_PER_BLOCK + (threadIdx.x >> 5);
    if (row >= batch) return;

    const v4f* __restrict__ rp = (const v4f*)(logits + (size_t)row * NUM_CLASS);
    const int target = y[row];              // uniform per wave -> scalar load

    const float NEG_INF = -__builtin_huge_valf();

    // ---- issue all 8 non-temporal b128 loads up front (MLP) ----
    v4f v[8];
#pragma unroll
    for (int k = 0; k < 7; ++k)
        v[k] = __builtin_nontemporal_load(rp + (k * 32 + lane));
    {
        const int i4 = 7 * 32 + lane;       // 224..255; valid iff < 250
        if (i4 < N_F4) {
            v[7] = __builtin_nontemporal_load(rp + i4);
        } else {
            v4f pad = {NEG_INF, NEG_INF, NEG_INF, NEG_INF};
            v[7] = pad;                     // -inf padding: exp()->0, fmax no-op
        }
    }

    // ---- per-lane local max over the 32 held elements ----
    float m = NEG_INF;
#pragma unroll
    for (int k = 0; k < 8; ++k)
        m = fmaxf(m, fmaxf(fmaxf(v[k][0], v[k][1]), fmaxf(v[k][2], v[k][3])));

    // ---- per-lane exp-sum + true-class logit + max wrong logit ----
    float s  = 0.0f;
    float mw = NEG_INF;                     // max logit over wrong classes
    float xy = NEG_INF;                     // logit of the true class
#pragma unroll
    for (int k = 0; k < 8; ++k) {
        const int colbase = (k * 32 + lane) * 4;   // un-clamped: pad cols >= 1000
#pragma unroll
        for (int c = 0; c < 4; ++c) {
            const float val = v[k][c];
            s += __expf(val - m);                  // pad: exp(-inf)=0
            const bool is_t = (colbase + c) == target;
            xy = is_t ? val : xy;
            mw = is_t ? mw  : fmaxf(mw, val);
        }
    }

    // ---- wave32 tree reduction (gfx1250 ds_swizzle/permlane path) ----
#pragma unroll
    for (int off = 16; off > 0; off >>= 1) {
        const float m2 = __shfl_xor(m, off, 32);
        const float s2 = __shfl_xor(s, off, 32);
        const float mn = fmaxf(m, m2);
        // every lane holds >= 28 real elements, so m, m2 are finite
        s = s * __expf(m - mn) + s2 * __expf(m2 - mn);
        m = mn;
        mw = fmaxf(mw, __shfl_xor(mw, off, 32));
        xy = fmaxf(xy, __shfl_xor(xy, off, 32));
    }

    if (lane == 0) {
        const float inv    = 1.0f / s;
        const float prob_y = __expf(xy - m) * inv;
        const float maxw   = __expf(mw - m) * inv;   // exp monotone: max prob == exp(max wrong logit)
        out[row] = 1.0f + maxw - prob_y;
    }
}

extern "C" void kernel_launch(void* const* d_in, const int* in_sizes, int n_in,
                              void* d_out, int out_size, void* d_ws, size_t ws_size,
                              hipStream_t stream) {
    const float* logits = (const float*)d_in[0];   // [B, 1000] fp32
    const int*   y      = (const int*)d_in[1];     // [B] int
    float*       out    = (float*)d_out;           // [B] fp32
    const int batch = out_size;                    // 262144

    const int blocks = (batch + WAVES_PER_BLOCK - 1) / WAVES_PER_BLOCK;
    DecisionBoundary_kernel<<<blocks, BLOCK, 0, stream>>>(logits, y, out, batch);
}